// MeshMamba3D_Seg_38371237822542
// MI455X (gfx1250) — compile-verified
//
#include <hip/hip_runtime.h>
#include <math.h>

// ---------------------------------------------------------------------------
// MeshMamba3D_Seg fused pipeline for MI455X (gfx1250, wave32).
// GEMMs run on V_WMMA_F32_16X16X4_F32 (exact f32; problem is only ~34 GFLOP,
// precision of LayerNorm/global-std matters more than peak matrix rate).
// ---------------------------------------------------------------------------

#define Bn 8
#define Gn 1024
#define Cn 384
#define C2n 768
#define EPSF 1e-5f

typedef __attribute__((ext_vector_type(2))) float v2f;
typedef __attribute__((ext_vector_type(8))) float v8f;

// ---------------- workspace layout (bytes) ---------------------------------
#define OFF_IDX     0UL                       // 8*1024*16 int      = 524288
#define OFF_STATS   524288UL                  // 32 doubles         = 256
#define OFF_STD     524544UL                  // 3 f32
#define OFF_FW      524560UL                  // 3 f32
#define OFF_FMEAN   524800UL                  // 8*384 f32          = 12288
#define OFF_FSTD    537088UL                  // 8*384 f32          = 12288
#define OFF_GANORM  549376UL                  // 8*1024*384 f32     = 12582912
#define OFF_LC      13132288UL                // 8*1024*768 f32     = 25165824
#define OFF_XCAT    38298112UL                // 8*2048*384 f32     = 25165824
// total ~63.5 MB

// ---------------------------------------------------------------------------
// 1) KNN: per row, top-16 smallest distances (stable ties -> lower index).
//    One batch's centers cached in LDS; 128 rows per block.
// ---------------------------------------------------------------------------
__global__ __launch_bounds__(128) void knn_kernel(const float* __restrict__ center,
                                                  int* __restrict__ idx_out) {
  int b = blockIdx.x >> 3;
  int nbase = (blockIdx.x & 7) * 128;
  __shared__ float sx[Gn], sy[Gn], sz[Gn], ssq[Gn];
  const float* cb = center + (size_t)b * Gn * 3;
  for (int m = threadIdx.x; m < Gn; m += blockDim.x) {
    float x = cb[m * 3 + 0], y = cb[m * 3 + 1], z = cb[m * 3 + 2];
    sx[m] = x; sy[m] = y; sz[m] = z; ssq[m] = x * x + y * y + z * z;
  }
  __syncthreads();
  int n = nbase + threadIdx.x;
  float cx = sx[n], cy = sy[n], cz = sz[n], sn = ssq[n];
  float bd[16]; int bi[16];
#pragma unroll
  for (int j = 0; j < 16; j++) { bd[j] = 3.4e38f; bi[j] = -1; }
  for (int m = 0; m < Gn; m++) {
    float d = sn + ssq[m] - 2.f * (cx * sx[m] + cy * sy[m] + cz * sz[m]);
    if (d < bd[15]) {
      int p = 15;
      while (p > 0 && bd[p - 1] > d) { bd[p] = bd[p - 1]; bi[p] = bi[p - 1]; p--; }
      bd[p] = d; bi[p] = m;
    }
  }
  int* o = idx_out + ((size_t)b * Gn + n) * 16;
#pragma unroll
  for (int j = 0; j < 16; j++) o[j] = bi[j];
}

// ---------------------------------------------------------------------------
// 2) Global std over (knn - center) diffs, per neighbor-rank partial sums so
//    one pass covers k = 4 / 8 / 16 via prefix sums.
// ---------------------------------------------------------------------------
__global__ void zero_stats_kernel(double* stats) {
  if (threadIdx.x < 32) stats[threadIdx.x] = 0.0;
}

__global__ __launch_bounds__(256) void diff_stats_kernel(const float* __restrict__ feat,
                                                         const int* __restrict__ idx,
                                                         double* __restrict__ stats) {
  int t = blockIdx.x * blockDim.x + threadIdx.x;       // [0, 8*1024*16)
  int b = t >> 14, g = (t >> 4) & 1023, j = t & 15;
  int nb = idx[((size_t)b * Gn + g) * 16 + j];
  const float* fc = feat + ((size_t)b * Gn + g) * Cn;
  const float* fn = feat + ((size_t)b * Gn + nb) * Cn;
  double s = 0.0, q = 0.0;
  for (int c = 0; c < Cn; c++) {
    double d = (double)fn[c] - (double)fc[c];
    s += d; q += d * d;
  }
  __shared__ double shs[16], shq[16];
  if (threadIdx.x < 16) { shs[threadIdx.x] = 0.0; shq[threadIdx.x] = 0.0; }
  __syncthreads();
  atomicAdd(&shs[j], s);
  atomicAdd(&shq[j], q);
  __syncthreads();
  if (threadIdx.x < 16) {
    atomicAdd(&stats[threadIdx.x], shs[threadIdx.x]);
    atomicAdd(&stats[16 + threadIdx.x], shq[threadIdx.x]);
  }
}

__global__ void finalize_stats_kernel(const double* __restrict__ stats,
                                      const float* __restrict__ fusion_w,
                                      float* __restrict__ std_out,
                                      float* __restrict__ w_out) {
  if (threadIdx.x == 0) {
    const int KG[3] = {4, 8, 16};
    for (int i = 0; i < 3; i++) {
      double s = 0.0, q = 0.0;
      for (int j = 0; j < KG[i]; j++) { s += stats[j]; q += stats[16 + j]; }
      double N = (double)Bn * Gn * KG[i] * Cn;
      double var = (q - s * s / N) / (N - 1.0);
      std_out[i] = (float)sqrt(var > 0.0 ? var : 0.0);
    }
    float m = fmaxf(fmaxf(fusion_w[0], fusion_w[1]), fusion_w[2]);
    float e0 = expf(fusion_w[0] - m), e1 = expf(fusion_w[1] - m), e2 = expf(fusion_w[2] - m);
    float den = e0 + e1 + e2;
    w_out[0] = e0 / den; w_out[1] = e1 / den; w_out[2] = e2 / den;
  }
}

// ---------------------------------------------------------------------------
// 3) feat per-(b,c) mean/std over G (ddof=1), then global-affine normalize.
// ---------------------------------------------------------------------------
__global__ void feat_stats_kernel(const float* __restrict__ feat,
                                  float* __restrict__ mean_out,
                                  float* __restrict__ std_out) {
  int t = blockIdx.x * blockDim.x + threadIdx.x;
  if (t >= Bn * Cn) return;
  int b = t / Cn, c = t % Cn;
  const float* f = feat + (size_t)b * Gn * Cn + c;
  double s = 0.0, q = 0.0;
  for (int g = 0; g < Gn; g++) { double v = f[(size_t)g * Cn]; s += v; q += v * v; }
  double mean = s / Gn;
  double var = (q - s * s / Gn) / (Gn - 1);
  mean_out[t] = (float)mean;
  std_out[t] = (float)sqrt(var > 0.0 ? var : 0.0);
}

__global__ void ga_norm_kernel(const float* __restrict__ feat,
                               const float* __restrict__ mean,
                               const float* __restrict__ stdv,
                               const float* __restrict__ alpha,
                               const float* __restrict__ beta,
                               float* __restrict__ out) {
  size_t t = (size_t)blockIdx.x * blockDim.x + threadIdx.x;
  if (t >= (size_t)Bn * Gn * Cn) return;
  int c = (int)(t % Cn);
  size_t bc = (t / ((size_t)Gn * Cn)) * Cn + c;
  out[t] = alpha[c] * ((feat[t] - mean[bc]) / (stdv[bc] + EPSF)) + beta[c];
}

// ---------------------------------------------------------------------------
// 4) Pooling (softmax-weighted over k) + LayerNorm(768) fused, per k-group.
//    One block per (b,g) row.
// ---------------------------------------------------------------------------
__global__ __launch_bounds__(256) void lc_kernel(const float* __restrict__ feat,
                                                 const int* __restrict__ idx,
                                                 const float* __restrict__ lnp_alpha,
                                                 const float* __restrict__ lnp_beta,
                                                 const float* __restrict__ ln_gamma,
                                                 const float* __restrict__ ln_betap,
                                                 const float* __restrict__ std_ws,
                                                 int grp, int k,
                                                 float* __restrict__ lc_out) {
  int bg = blockIdx.x;
  int b = bg >> 10;
  __shared__ float lcbuf[C2n];
  __shared__ int nIdx[16];
  __shared__ float rs[256], rq[256];
  if (threadIdx.x < 16) nIdx[threadIdx.x] = idx[(size_t)bg * 16 + threadIdx.x];
  __syncthreads();
  float invstd = 1.f / (std_ws[grp] + EPSF);
  const float* al = lnp_alpha + grp * C2n;
  const float* be = lnp_beta + grp * C2n;
  const float* frow = feat + (size_t)bg * Cn;
  float ps = 0.f, pq = 0.f;
  for (int d = threadIdx.x; d < C2n; d += blockDim.x) {
    float lcd;
    if (d < Cn) {
      float fc = frow[d];
      float se = 0.f, sxe = 0.f;
      for (int j = 0; j < k; j++) {
        float x = al[d] * ((feat[((size_t)b * Gn + nIdx[j]) * Cn + d] - fc) * invstd) + be[d];
        float e = expf(x);
        se += e; sxe += x * e;
      }
      lcd = sxe / se;
    } else {
      lcd = al[d] * frow[d - Cn] + be[d];   // mean branch: all k identical
    }
    lcbuf[d] = lcd; ps += lcd; pq += lcd * lcd;
  }
  rs[threadIdx.x] = ps; rq[threadIdx.x] = pq;
  __syncthreads();
  for (int s = 128; s > 0; s >>= 1) {
    if (threadIdx.x < s) { rs[threadIdx.x] += rs[threadIdx.x + s]; rq[threadIdx.x] += rq[threadIdx.x + s]; }
    __syncthreads();
  }
  float mu = rs[0] / C2n;
  float var = rq[0] / C2n - mu * mu;
  float rstd = rsqrtf(var + EPSF);
  const float* gm = ln_gamma + grp * C2n;
  const float* bt = ln_betap + grp * C2n;
  float* orow = lc_out + (size_t)bg * C2n;
  for (int d = threadIdx.x; d < C2n; d += blockDim.x)
    orow[d] = (lcbuf[d] - mu) * rstd * gm[d] + bt[d];
}

// ---------------------------------------------------------------------------
// 5) Generic f32 WMMA GEMM: Out[m,n] = sum_k A[m,k]*B[k,n] (+bias, epilogue).
//    64x64 block tile, K staged 32 at a time through LDS, 8 waves each doing
//    two 16x16 accumulators via V_WMMA_F32_16X16X4_F32.
//    A: row-major, contiguous in k (stride aRS per row, batch stride aBatch).
//    B: arbitrary strides bKS (k) / bNS (n), batch stride bBatch.
//    mode 0: Out (+)= scale * silu(acc + biasN[n])      (mlp fusion)
//    mode 1: Out  =  silu(acc + biasM[m])               (gaf)
//    mode 2: Out  =  acc + biasM[m]                     (final)
//    Requires M%64==0, N%64==0, K%32==0 (true for all calls here).
// ---------------------------------------------------------------------------
__global__ __launch_bounds__(256) void gemm_wmma_f32(
    const float* __restrict__ A, long aBatch, long aRS,
    const float* __restrict__ Bm, long bBatch, long bKS, long bNS,
    float* __restrict__ Out, long oBatch,
    const float* __restrict__ biasN, const float* __restrict__ biasM,
    int M, int N, int K, int mode,
    const float* __restrict__ scalePtr, int scaleIdx, int accumulate) {
  const int bz = blockIdx.z;
  const int mTile = blockIdx.y * 64;
  const int nTile = blockIdx.x * 64;
  __shared__ float As[64][36];
  __shared__ float Bs[32][68];
  const int tid = threadIdx.x;
  const int lane = tid & 31, wid = tid >> 5;
  const int mBase = (wid & 3) * 16;
  const int nBase = (wid >> 2) * 32;
  const int khalf = lane >> 4, lr = lane & 15;
  v8f acc0 = {};
  v8f acc1 = {};
  const float* Ab = A + (size_t)aBatch * bz;
  const float* Bb = Bm + (size_t)bBatch * bz;
  for (int k0 = 0; k0 < K; k0 += 32) {
#pragma unroll
    for (int e = tid; e < 2048; e += 256) {          // A tile 64x32
      int r = e >> 5, kk = e & 31;
      As[r][kk] = Ab[(size_t)(mTile + r) * aRS + (k0 + kk)];
    }
#pragma unroll
    for (int e = tid; e < 2048; e += 256) {          // B tile 32x64
      int kk = e >> 6, nn = e & 63;
      Bs[kk][nn] = Bb[(size_t)(k0 + kk) * bKS + (size_t)(nTile + nn) * bNS];
    }
    __syncthreads();
#pragma unroll
    for (int kk = 0; kk < 32; kk += 4) {
      // A fragment 16x4: lane L holds (m = L%16, k = v + 2*(L/16)), v = vgpr
      v2f a;
      a.x = As[mBase + lr][kk + 2 * khalf];
      a.y = As[mBase + lr][kk + 2 * khalf + 1];
      // B fragment 4x16: lane L holds (n = L%16, k = v + 2*(L/16))
      v2f b0, b1;
      b0.x = Bs[kk + 2 * khalf][nBase + lr];
      b0.y = Bs[kk + 2 * khalf + 1][nBase + lr];
      b1.x = Bs[kk + 2 * khalf][nBase + 16 + lr];
      b1.y = Bs[kk + 2 * khalf + 1][nBase + 16 + lr];
      acc0 = __builtin_amdgcn_wmma_f32_16x16x4_f32(false, a, false, b0, (short)0, acc0, false, false);
      acc1 = __builtin_amdgcn_wmma_f32_16x16x4_f32(false, a, false, b1, (short)0, acc1, false, false);
    }
    __syncthreads();
  }
  float scale = scalePtr ? scalePtr[scaleIdx] : 1.0f;
  float* Ob = Out + (size_t)oBatch * bz;
#pragma unroll
  for (int r = 0; r < 8; r++) {
    int m = mTile + mBase + r + 8 * khalf;    // D layout: lanes16-31 -> m+8
#pragma unroll
    for (int s = 0; s < 2; s++) {
      int n = nTile + nBase + s * 16 + lr;
      float v = (s == 0) ? acc0[r] : acc1[r];
      if (biasN) v += biasN[n];
      if (biasM) v += biasM[m];
      float o;
      if (mode == 2) {
        o = v;
      } else {
        float sg = v / (1.f + expf(-v));      // silu
        o = (mode == 0) ? sg * scale : sg;
      }
      size_t off = (size_t)m * N + n;
      if (mode == 0 && accumulate) Ob[off] += o;
      else Ob[off] = o;
    }
  }
}

// ---------------------------------------------------------------------------
extern "C" void kernel_launch(void* const* d_in, const int* in_sizes, int n_in,
                              void* d_out, int out_size, void* d_ws, size_t ws_size,
                              hipStream_t stream) {
  const float* center    = (const float*)d_in[0];
  const float* feat      = (const float*)d_in[1];
  const float* lnp_alpha = (const float*)d_in[2];
  const float* lnp_beta  = (const float*)d_in[3];
  const float* ln_gamma  = (const float*)d_in[4];
  const float* ln_beta   = (const float*)d_in[5];
  const float* mlp_w     = (const float*)d_in[6];
  const float* mlp_b     = (const float*)d_in[7];
  const float* fusion_w  = (const float*)d_in[8];
  const float* gaf_alpha = (const float*)d_in[9];
  const float* gaf_beta  = (const float*)d_in[10];
  const float* gaf_w     = (const float*)d_in[11];
  const float* gaf_b     = (const float*)d_in[12];
  const float* fin_w     = (const float*)d_in[13];
  const float* fin_b     = (const float*)d_in[14];
  float* out = (float*)d_out;

  char* ws = (char*)d_ws;
  int*    idx    = (int*)   (ws + OFF_IDX);
  double* stats  = (double*)(ws + OFF_STATS);
  float*  stdg   = (float*) (ws + OFF_STD);
  float*  fwts   = (float*) (ws + OFF_FW);
  float*  fmean  = (float*) (ws + OFF_FMEAN);
  float*  fstd   = (float*) (ws + OFF_FSTD);
  float*  ganorm = (float*) (ws + OFF_GANORM);
  float*  lcbuf  = (float*) (ws + OFF_LC);
  float*  xcat   = (float*) (ws + OFF_XCAT);   // (B, 2048, 384): [:1024]=ga, [1024:]=lnp

  // --- stage 1: KNN + global diff std + fusion weights --------------------
  knn_kernel<<<64, 128, 0, stream>>>(center, idx);
  zero_stats_kernel<<<1, 32, 0, stream>>>(stats);
  diff_stats_kernel<<<512, 256, 0, stream>>>(feat, idx, stats);
  finalize_stats_kernel<<<1, 32, 0, stream>>>(stats, fusion_w, stdg, fwts);

  // --- stage 2: global affine + gaf GEMM (silu) into xcat[:, :1024] -------
  feat_stats_kernel<<<12, 256, 0, stream>>>(feat, fmean, fstd);
  ga_norm_kernel<<<(Bn * Gn * Cn) / 256, 256, 0, stream>>>(feat, fmean, fstd,
                                                           gaf_alpha, gaf_beta, ganorm);
  gemm_wmma_f32<<<dim3(Cn / 64, Gn / 64, Bn), 256, 0, stream>>>(
      gaf_w, 0, Gn,                      // A = gaf_w (1024x1024), shared
      ganorm, (long)Gn * Cn, Cn, 1,      // B = ganorm[b] (k=g stride 384, n=c stride 1)
      xcat, 2048L * Cn,                  // Out = xcat[b][:1024]
      nullptr, gaf_b, Gn, Cn, Gn, /*mode=*/1, nullptr, 0, 0);

  // --- stage 3: per-k pooling+LN, MLP GEMM fused silu+fusion into lnp -----
  const int KG[3] = {4, 8, 16};
  for (int i = 0; i < 3; i++) {
    lc_kernel<<<Bn * Gn, 256, 0, stream>>>(feat, idx, lnp_alpha, lnp_beta,
                                           ln_gamma, ln_beta, stdg, i, KG[i], lcbuf);
    gemm_wmma_f32<<<dim3(Cn / 64, Gn / 64, Bn), 256, 0, stream>>>(
        lcbuf, (long)Gn * C2n, C2n,                 // A = lc[b] (1024x768)
        mlp_w + (size_t)i * Cn * C2n, 0, 1, C2n,    // B[k,n] = mlp_w[i][n,k]
        xcat + (size_t)Gn * Cn, 2048L * Cn,         // Out = xcat[b][1024:]
        mlp_b + i * Cn, nullptr, Gn, Cn, C2n,
        /*mode=*/0, fwts, i, /*accumulate=*/(i > 0) ? 1 : 0);
  }

  // --- stage 4: final GEMM over concatenated xcat -------------------------
  gemm_wmma_f32<<<dim3(Cn / 64, Gn / 64, Bn), 256, 0, stream>>>(
      fin_w, 0, 2L * Gn,                 // A = fin_w (1024x2048), shared
      xcat, 2048L * Cn, Cn, 1,           // B = xcat[b] (2048x384)
      out, (long)Gn * Cn,
      nullptr, fin_b, Gn, Cn, 2 * Gn, /*mode=*/2, nullptr, 0, 0);
}